// Encoder_57483842290158
// MI455X (gfx1250) — compile-verified
//
#include <hip/hip_runtime.h>
#include <hip/hip_bf16.h>

#define VOCAB 30000
#define H     1024
#define H3    3072
#define BATCH 64
#define SEQ   512

typedef __attribute__((ext_vector_type(2))) float v2f;
typedef __attribute__((ext_vector_type(8))) float v8f;

// V_WMMA_F32_16X16X4_F32: D(16x16 f32, 8 VGPR) = A(16x4 f32, 2 VGPR) x B(4x16 f32, 2 VGPR) + C
__device__ __forceinline__ v8f wmma_f32(v2f a, v2f b, v8f c) {
    return __builtin_amdgcn_wmma_f32_16x16x4_f32(
        /*neg_a=*/false, a, /*neg_b=*/false, b,
        /*c_mod=*/(short)0, c, /*reuse_a=*/false, /*reuse_b=*/false);
}

__device__ __forceinline__ float sigmoidf_(float x) {
    return 1.0f / (1.0f + __expf(-x));
}

// A-tile pitch (floats): reads go down an M-column; 36 spreads the 16 lanes of
// each half-wave over distinct banks (lane halves land on banks =0,1 vs =2,3 mod 4).
#define APAD 36
// B tiles are stored k-pair interleaved: element (k,n) -> word (k>>1)*PITCH + 2n + (k&1),
// so a B fragment (B[k][n], B[k+1][n]) is ONE aligned ds_load_b64.
// PITCH % 64 == 32 puts the two lane-halves (pair-rows p and p+1) in disjoint
// 32-bank ranges -> conflict-free.
#define BP1 160   // phase-1 B tile: 16 pair-rows x (64 cols * 2 + 32 pad)
#define BP2  96   // phase-2 U tile: 16 pair-rows x (32 cols * 2 + 32 pad)

// ---------------------------------------------------------------------------
// Phase 1: x_proj[r][c] = emb_table[tokens[r]][:] @ W[:, c] + b_i[c]
// Grid: (BATCH*SEQ/64, H3/64), 256 threads (8 waves). Each WG: 64x64 output.
// ---------------------------------------------------------------------------
__global__ __launch_bounds__(256) void xproj_kernel(
    const int*   __restrict__ tokens,   // [B*S]
    const float* __restrict__ emb,      // [VOCAB][H]
    const float* __restrict__ W,        // [H][H3]
    const float* __restrict__ b_i,      // [H3]
    float*       __restrict__ xproj)    // [B*S][H3]
{
    __shared__ float As[64 * APAD];     // 64 rows x 32 K (padded)
    __shared__ float Bs[16 * BP1];      // 32 K x 64 N (k-pair interleaved)
    __shared__ int   stok[64];

    const int tid  = threadIdx.x;
    const int lane = tid & 31;
    const int wave = tid >> 5;
    const int r0 = blockIdx.x * 64;     // output row base
    const int c0 = blockIdx.y * 64;     // output col base

    if (tid < 64) stok[tid] = tokens[r0 + tid];
    __syncthreads();

    const int mtile = wave & 3;         // 0..3
    const int npair = wave >> 2;        // 0..1
    v8f acc0 = {}; v8f acc1 = {};

    const int mrow  = mtile * 16 + (lane & 15);
    const int khalf = (lane >> 4) * 2;          // 0 or 2
    const int ncol  = npair * 32 + (lane & 15); // col-in-64 for tile 0; tile1 = +16

    for (int k0 = 0; k0 < H; k0 += 32) {
        __syncthreads();
        // A: gathered emb rows, 64x32 floats = 512 float4 slots
        for (int s = tid; s < 512; s += 256) {
            int row = s >> 3;
            int c4  = (s & 7) * 4;
            const float* src = emb + (size_t)stok[row] * H + k0 + c4;
            float4 v = *(const float4*)src;
            *(float4*)&As[row * APAD + c4] = v;
            if (k0 + 32 < H) __builtin_prefetch(src + 32, 0, 1);   // next K-chunk
        }
        // B: W tile, 32x64 floats -> k-pair interleaved scatter
        for (int s = tid; s < 512; s += 256) {
            int k   = s >> 4;
            int c4  = (s & 15) * 4;
            const float* src = W + (size_t)k * H3 + k0 * H3 / 32 * 0 + (size_t)(k0 + 0) * 0; // (see below)
            (void)src;
            const float* wsrc = W + (size_t)(k0 + k) * H3 + c0 + c4;
            float4 v = *(const float4*)wsrc;
            float* dst = &Bs[(k >> 1) * BP1 + (k & 1)];
            dst[2 * (c4 + 0)] = v.x;
            dst[2 * (c4 + 1)] = v.y;
            dst[2 * (c4 + 2)] = v.z;
            dst[2 * (c4 + 3)] = v.w;
            if (k0 + 32 < H) __builtin_prefetch(wsrc + 32 * H3, 0, 1);  // next K-chunk
        }
        __syncthreads();
        #pragma unroll
        for (int kk = 0; kk < 32; kk += 4) {
            v2f a  = *(const v2f*)&As[mrow * APAD + kk + khalf];
            const int prow = ((kk + khalf) >> 1) * BP1;
            v2f b0 = *(const v2f*)&Bs[prow + 2 * ncol];
            v2f b1 = *(const v2f*)&Bs[prow + 2 * (ncol + 16)];
            acc0 = wmma_f32(a, b0, acc0);
            acc1 = wmma_f32(a, b1, acc1);
        }
    }

    const int gcol = c0 + ncol;
    const float bi0 = b_i[gcol];
    const float bi1 = b_i[gcol + 16];
    const int mbase = r0 + mtile * 16 + (lane >> 4) * 8;
    #pragma unroll
    for (int v = 0; v < 8; ++v) {
        size_t row = (size_t)(mbase + v);
        xproj[row * H3 + gcol]      = acc0[v] + bi0;
        xproj[row * H3 + gcol + 16] = acc1[v] + bi1;
    }
}

// ---------------------------------------------------------------------------
// Phase 2 (per time step t): rp = h_in @ U + b_r (3 gates), gate math,
// masked update, write hs[:,t,:] and h_out.
// Grid: (H/32) WGs, 256 threads (8 waves). Each WG: 64 batches x 32 h-cols.
// ---------------------------------------------------------------------------
__global__ __launch_bounds__(256) void gru_step_kernel(
    const int*   __restrict__ tokens,   // [B*S]
    const float* __restrict__ xproj,    // [B*S][H3]
    const float* __restrict__ U,        // [H][H3]
    const float* __restrict__ b_r,      // [H3]
    const float* __restrict__ h_in,     // [B][H]
    float*       __restrict__ h_out,    // [B][H]
    float*       __restrict__ hs,       // [B][S][H]
    float*       __restrict__ h_last,   // [B][H]
    int t)
{
    __shared__ float Hs[64 * APAD];         // 64 batches x 32 K (padded)
    __shared__ float Us[3][16 * BP2];       // per gate: 32 K x 32 N (k-pair interleaved)

    const int tid  = threadIdx.x;
    const int lane = tid & 31;
    const int wave = tid >> 5;
    const int c0 = blockIdx.x * 32;         // column base within H

    const int mtile = wave & 3;             // batch tile 0..3
    const int ntile = wave >> 2;            // 16-col tile 0..1

    v8f accz = {}; v8f accr = {}; v8f acch = {};

    const int mrow  = mtile * 16 + (lane & 15);
    const int khalf = (lane >> 4) * 2;
    const int ncol  = ntile * 16 + (lane & 15);   // 0..31 within block

    for (int k0 = 0; k0 < H; k0 += 32) {
        __syncthreads();
        // h tile: 64x32 floats = 512 float4 slots
        for (int s = tid; s < 512; s += 256) {
            int row = s >> 3;
            int c4  = (s & 7) * 4;
            float4 v = *(const float4*)(h_in + (size_t)row * H + k0 + c4);
            *(float4*)&Hs[row * APAD + c4] = v;
        }
        // U tiles for 3 gates: 3 x (32x32) floats -> k-pair interleaved scatter
        for (int s = tid; s < 768; s += 256) {
            int g  = s >> 8;           // gate 0..2
            int k  = (s >> 3) & 31;    // K row 0..31
            int c4 = (s & 7) * 4;      // col 0..28
            float4 v = *(const float4*)(U + (size_t)(k0 + k) * H3 + g * H + c0 + c4);
            float* dst = &Us[g][(k >> 1) * BP2 + (k & 1)];
            dst[2 * (c4 + 0)] = v.x;
            dst[2 * (c4 + 1)] = v.y;
            dst[2 * (c4 + 2)] = v.z;
            dst[2 * (c4 + 3)] = v.w;
        }
        __syncthreads();
        #pragma unroll
        for (int kk = 0; kk < 32; kk += 4) {
            v2f a = *(const v2f*)&Hs[mrow * APAD + kk + khalf];
            const int prow = ((kk + khalf) >> 1) * BP2;
            v2f bz = *(const v2f*)&Us[0][prow + 2 * ncol];
            v2f br = *(const v2f*)&Us[1][prow + 2 * ncol];
            v2f bh = *(const v2f*)&Us[2][prow + 2 * ncol];
            accz = wmma_f32(a, bz, accz);
            accr = wmma_f32(a, br, accr);
            acch = wmma_f32(a, bh, acch);
        }
    }

    const int j = c0 + ncol;               // global column in H
    const float brz = b_r[j];
    const float brr = b_r[H + j];
    const float brh = b_r[2 * H + j];
    const int bbase = mtile * 16 + (lane >> 4) * 8;
    #pragma unroll
    for (int v = 0; v < 8; ++v) {
        int batch = bbase + v;
        size_t xrow = (size_t)(batch * SEQ + t) * H3;
        float xz = xproj[xrow + j];
        float xr = xproj[xrow + H + j];
        float xh = xproj[xrow + 2 * H + j];
        float rz = accz[v] + brz;
        float rr = accr[v] + brr;
        float rh = acch[v] + brh;
        float z  = sigmoidf_(xz + rz);
        float r  = sigmoidf_(xr + rr);
        float hh = tanhf(xh + r * rh);
        float hp = h_in[(size_t)batch * H + j];
        float hn = z * hp + (1.0f - z) * hh;
        if (tokens[batch * SEQ + t] == 0) hn = hp;      // mask
        h_out[(size_t)batch * H + j] = hn;
        hs[((size_t)batch * SEQ + t) * H + j] = hn;
        if (t == SEQ - 1) h_last[(size_t)batch * H + j] = hn;
    }
}

__global__ void zero_kernel(float* __restrict__ p, int n) {
    int i = blockIdx.x * blockDim.x + threadIdx.x;
    if (i < n) p[i] = 0.0f;
}

extern "C" void kernel_launch(void* const* d_in, const int* in_sizes, int n_in,
                              void* d_out, int out_size, void* d_ws, size_t ws_size,
                              hipStream_t stream) {
    const int*   tokens = (const int*)  d_in[0];
    const float* emb    = (const float*)d_in[1];
    const float* W      = (const float*)d_in[2];
    const float* U      = (const float*)d_in[3];
    const float* b_i    = (const float*)d_in[4];
    const float* b_r    = (const float*)d_in[5];

    float* hs     = (float*)d_out;                          // [B][S][H]
    float* h_last = hs + (size_t)BATCH * SEQ * H;           // [B][H]

    // Workspace: xproj (B*S*3H floats = 402 MB) + two h ping-pong buffers.
    float* xproj = (float*)d_ws;
    float* hbuf0 = xproj + (size_t)BATCH * SEQ * H3;
    float* hbuf1 = hbuf0 + (size_t)BATCH * H;

    // h0 = 0 (must re-init every launch; harness doesn't restore d_ws)
    zero_kernel<<<dim3((BATCH * H + 255) / 256), dim3(256), 0, stream>>>(hbuf0, BATCH * H);

    // Phase 1: fused gather + GEMM + bias
    xproj_kernel<<<dim3((BATCH * SEQ) / 64, H3 / 64), dim3(256), 0, stream>>>(
        tokens, emb, W, b_i, xproj);

    // Phase 2: 512 stream-ordered fused GRU steps (ping-pong h buffers)
    for (int t = 0; t < SEQ; ++t) {
        const float* hin = (t & 1) ? hbuf1 : hbuf0;
        float*       hout = (t & 1) ? hbuf0 : hbuf1;
        gru_step_kernel<<<dim3(H / 32), dim3(256), 0, stream>>>(
            tokens, xproj, U, b_r, hin, hout, hs, h_last, t);
    }
}